// TransitionUp_7645041787059
// MI455X (gfx1250) — compile-verified
//
#include <hip/hip_runtime.h>

typedef __attribute__((ext_vector_type(2))) float v2f;
typedef __attribute__((ext_vector_type(8))) float v8f;

#define BATCH 4
#define ND    4096
#define NU    16384
#define CIN   256
#define COUT  128
#define KNN   4
#define BN_EPS 1e-5f
#define EPS_D2 1e-16f

// ---------------------------------------------------------------------------
// Kernel 1: xd = ReLU(BN(x_down @ W_down))    [B*ND, CIN] x [CIN, COUT]
// One block = 16 output rows x 128 cols. 8 waves, each wave owns a 16-col
// slice and runs 64 x V_WMMA_F32_16X16X4_F32 over K=256.
// A tile staged in LDS (16x256 f32 = 16KB).
// ---------------------------------------------------------------------------
__global__ __launch_bounds__(256, 2) void mlp_down_kernel(
    const float* __restrict__ x, const float* __restrict__ W,
    const float* __restrict__ bias, const float* __restrict__ gam,
    const float* __restrict__ bet, const float* __restrict__ mu,
    const float* __restrict__ var, float* __restrict__ xd)
{
    __shared__ __align__(16) float As[16 * CIN];
    const int tid = threadIdx.x;
    const long m0 = (long)blockIdx.x * 16;

    // Stage A tile: 16 rows x 256 f32 = 1024 float4; 4 per thread.
    {
        const float4* src = (const float4*)(x + m0 * CIN);
        float4* dst = (float4*)As;
#pragma unroll
        for (int i = 0; i < 4; ++i)
            dst[tid + 256 * i] = src[tid + 256 * i];
    }
    __syncthreads();

    const int wave   = tid >> 5;
    const int lane   = tid & 31;
    const int laneLo = lane & 15;
    const int hi     = lane >> 4;       // 0: K lanes 0/1, 1: K lanes 2/3
    const int c0     = wave * 16;
    const int ch     = c0 + laneLo;     // output channel this lane owns

    const float* Arow = As + laneLo * CIN;   // A rows M=0..15 live in laneLo
    v8f acc = {};

    __builtin_prefetch(&W[c0], 0, 1);
#pragma unroll 8
    for (int k = 0; k < CIN; k += 4) {
        v2f a, b;
        // fp32 A 16x4 layout: v0 = K {0|2}, v1 = K {1|3} (low|high lanes)
        a[0] = Arow[k + 2 * hi + 0];
        a[1] = Arow[k + 2 * hi + 1];
        // B 4x16: rows striped across lanes, same K split as A
        b[0] = W[(long)(k + 2 * hi + 0) * COUT + ch];
        b[1] = W[(long)(k + 2 * hi + 1) * COUT + ch];
        acc = __builtin_amdgcn_wmma_f32_16x16x4_f32(false, a, false, b,
                                                    (short)0, acc, false, false);
    }

    // Fused bias + BN(eval) + ReLU epilogue.
    const float bb    = bias[ch];
    const float scale = gam[ch] * rsqrtf(var[ch] + BN_EPS);
    const float mm    = mu[ch];
    const float bee   = bet[ch];
#pragma unroll
    for (int r = 0; r < 8; ++r) {
        const long row = m0 + r + 8 * hi;    // C/D layout: VGPR r -> M = r + 8*hi
        float z = acc[r] + bb;
        z = scale * (z - mm) + bee;
        xd[row * COUT + ch] = fmaxf(z, 0.f);
    }
}

// ---------------------------------------------------------------------------
// Kernel 2: kNN(4) over 4096 down points per up point.
// All down coords staged SoA in LDS (48KB); every lane scans with broadcast
// LDS reads and keeps a sorted top-4 (predicated bubble insert).
// Emits normalized inverse-squared-distance weights.
// ---------------------------------------------------------------------------
__global__ __launch_bounds__(256) void knn_kernel(
    const float* __restrict__ p_down, const float* __restrict__ p_up,
    int* __restrict__ idx_out, float* __restrict__ w_out)
{
    __shared__ float pdx[ND];
    __shared__ float pdy[ND];
    __shared__ float pdz[ND];

    const int b = blockIdx.y;
    const float* pd = p_down + (long)b * ND * 3;
    for (int i = threadIdx.x; i < ND; i += 256) {
        pdx[i] = pd[i * 3 + 0];
        pdy[i] = pd[i * 3 + 1];
        pdz[i] = pd[i * 3 + 2];
    }
    __syncthreads();

    const long u = (long)blockIdx.x * 256 + threadIdx.x;   // up point in batch
    const float* pu = p_up + ((long)b * NU + u) * 3;
    const float qx = pu[0], qy = pu[1], qz = pu[2];

    float d0 = 3.4e38f, d1 = 3.4e38f, d2 = 3.4e38f, d3 = 3.4e38f;
    int   i0 = 0, i1 = 0, i2 = 0, i3 = 0;

#pragma unroll 4
    for (int j = 0; j < ND; ++j) {
        const float dx = qx - pdx[j];
        const float dy = qy - pdy[j];
        const float dz = qz - pdz[j];
        const float d = dx * dx + dy * dy + dz * dz;
        if (d < d3) {
            d3 = d; i3 = j;
            if (d3 < d2) {
                float td = d3; d3 = d2; d2 = td; int ti = i3; i3 = i2; i2 = ti;
                if (d2 < d1) {
                    td = d2; d2 = d1; d1 = td; ti = i2; i2 = i1; i1 = ti;
                    if (d1 < d0) {
                        td = d1; d1 = d0; d0 = td; ti = i1; i1 = i0; i0 = ti;
                    }
                }
            }
        }
    }

    const float w0 = 1.f / fmaxf(d0, EPS_D2);
    const float w1 = 1.f / fmaxf(d1, EPS_D2);
    const float w2 = 1.f / fmaxf(d2, EPS_D2);
    const float w3 = 1.f / fmaxf(d3, EPS_D2);
    const float inv = 1.f / (w0 + w1 + w2 + w3);

    const long base = ((long)b * NU + u) * KNN;
    idx_out[base + 0] = i0; idx_out[base + 1] = i1;
    idx_out[base + 2] = i2; idx_out[base + 3] = i3;
    w_out[base + 0] = w0 * inv; w_out[base + 1] = w1 * inv;
    w_out[base + 2] = w2 * inv; w_out[base + 3] = w3 * inv;
}

// ---------------------------------------------------------------------------
// Kernel 3: out = knn_interp(xd) + ReLU(BN(x_up @ W_up))
// 16 rows x 128 cols per block. Gather/weighted-sum builds an LDS interp
// tile while the x_up tile is also staged; 8 waves then run the K=128 fp32
// WMMA GEMM and add the interp tile in the epilogue.
// ---------------------------------------------------------------------------
__global__ __launch_bounds__(256, 2) void fuse_up_kernel(
    const float* __restrict__ xu, const float* __restrict__ W,
    const float* __restrict__ bias, const float* __restrict__ gam,
    const float* __restrict__ bet, const float* __restrict__ mu,
    const float* __restrict__ var,
    const float* __restrict__ xd, const int* __restrict__ knn_idx,
    const float* __restrict__ knn_w, float* __restrict__ out)
{
    __shared__ __align__(16) float As[16 * COUT];      // x_up tile (8KB)
    __shared__ __align__(16) float Interp[16 * COUT];  // interp tile (8KB)

    const int tid = threadIdx.x;
    const long m0 = (long)blockIdx.x * 16;   // row tile over B*NU
    const int  b  = (int)(m0 / NU);          // NU % 16 == 0: no straddle

    // Stage A tile: 16x128 f32 = 512 float4; 2 per thread.
    {
        const float4* src = (const float4*)(xu + m0 * COUT);
        float4* dst = (float4*)As;
        dst[tid]       = src[tid];
        dst[tid + 256] = src[tid + 256];
    }

    // Gather + weighted sum into Interp: thread owns (row r, 8 channels).
    {
        const int r  = tid >> 4;
        const int cb = (tid & 15) * 8;
        const long row = m0 + r;                    // == b*NU + u
        const int*   id = knn_idx + row * KNN;
        const float* wv = knn_w   + row * KNN;
        float4 aA = {0.f, 0.f, 0.f, 0.f};
        float4 aB = {0.f, 0.f, 0.f, 0.f};
#pragma unroll
        for (int k = 0; k < KNN; ++k) {
            const float wk = wv[k];
            const float* f = xd + ((long)b * ND + id[k]) * COUT + cb;
            const float4 f0 = *(const float4*)f;
            const float4 f1 = *(const float4*)(f + 4);
            aA.x += wk * f0.x; aA.y += wk * f0.y;
            aA.z += wk * f0.z; aA.w += wk * f0.w;
            aB.x += wk * f1.x; aB.y += wk * f1.y;
            aB.z += wk * f1.z; aB.w += wk * f1.w;
        }
        *(float4*)&Interp[r * COUT + cb]     = aA;
        *(float4*)&Interp[r * COUT + cb + 4] = aB;
    }
    __syncthreads();

    const int wave   = tid >> 5;
    const int lane   = tid & 31;
    const int laneLo = lane & 15;
    const int hi     = lane >> 4;
    const int c0     = wave * 16;
    const int ch     = c0 + laneLo;

    const float* Arow = As + laneLo * COUT;
    v8f acc = {};

    __builtin_prefetch(&W[c0], 0, 1);
#pragma unroll 8
    for (int k = 0; k < COUT; k += 4) {
        v2f a, bf;
        a[0]  = Arow[k + 2 * hi + 0];
        a[1]  = Arow[k + 2 * hi + 1];
        bf[0] = W[(long)(k + 2 * hi + 0) * COUT + ch];
        bf[1] = W[(long)(k + 2 * hi + 1) * COUT + ch];
        acc = __builtin_amdgcn_wmma_f32_16x16x4_f32(false, a, false, bf,
                                                    (short)0, acc, false, false);
    }

    const float bb    = bias[ch];
    const float scale = gam[ch] * rsqrtf(var[ch] + BN_EPS);
    const float mm    = mu[ch];
    const float bee   = bet[ch];
#pragma unroll
    for (int r = 0; r < 8; ++r) {
        const int rr = r + 8 * hi;               // row within tile
        float z = acc[r] + bb;
        z = scale * (z - mm) + bee;
        z = fmaxf(z, 0.f) + Interp[rr * COUT + ch];
        out[(m0 + rr) * COUT + ch] = z;
    }
}

// ---------------------------------------------------------------------------
extern "C" void kernel_launch(void* const* d_in, const int* in_sizes, int n_in,
                              void* d_out, int out_size, void* d_ws, size_t ws_size,
                              hipStream_t stream)
{
    const float* x_down  = (const float*)d_in[0];
    const float* x_up    = (const float*)d_in[1];
    const float* p_down  = (const float*)d_in[2];
    const float* p_up    = (const float*)d_in[3];
    const float* W_down  = (const float*)d_in[4];
    const float* b_down  = (const float*)d_in[5];
    const float* g_down  = (const float*)d_in[6];
    const float* be_down = (const float*)d_in[7];
    const float* m_down  = (const float*)d_in[8];
    const float* v_down  = (const float*)d_in[9];
    const float* W_up    = (const float*)d_in[10];
    const float* b_up    = (const float*)d_in[11];
    const float* g_up    = (const float*)d_in[12];
    const float* be_up   = (const float*)d_in[13];
    const float* m_up    = (const float*)d_in[14];
    const float* v_up    = (const float*)d_in[15];
    float* out = (float*)d_out;

    // Workspace layout: xd | knn_idx | knn_w  (~10.5 MB total)
    const size_t xd_bytes  = (size_t)BATCH * ND * COUT * sizeof(float);
    const size_t idx_bytes = (size_t)BATCH * NU * KNN * sizeof(int);
    float* xd      = (float*)d_ws;
    int*   knn_idx = (int*)((char*)d_ws + xd_bytes);
    float* knn_w   = (float*)((char*)d_ws + xd_bytes + idx_bytes);

    mlp_down_kernel<<<(BATCH * ND) / 16, 256, 0, stream>>>(
        x_down, W_down, b_down, g_down, be_down, m_down, v_down, xd);

    knn_kernel<<<dim3(NU / 256, BATCH), 256, 0, stream>>>(
        p_down, p_up, knn_idx, knn_w);

    fuse_up_kernel<<<(BATCH * NU) / 16, 256, 0, stream>>>(
        x_up, W_up, b_up, g_up, be_up, m_up, v_up, xd, knn_idx, knn_w, out);
}